// LocalCoherence_48180943126735
// MI455X (gfx1250) — compile-verified
//
#include <hip/hip_runtime.h>
#include <cmath>

// ---------------------------------------------------------------------------
// Local coherence map:  5 separable Gaussian smoothings (101-tap T, 11-tap X)
// of x, y, x^2, y^2, xy over [16,1,2500,512] fp32, then
// cs = cov * rsqrt(var_x*var_y + eps), output = mean(1 - cs)  (scalar).
//
// The 101-tap T-convolution (~90% of FLOPs) runs on the matrix pipes as a
// banded matmul with V_WMMA_F32_16X16X4_F32:
//   out[m,n] = sum_r A[m,r] * in[tb-50+r, n],   A[m,r] = k_t[r-m]  (16x116)
// K = 116 = 29 chunks of 4. One weight A-fragment is shared by all 5 field
// WMMAs per chunk. Interior row-blocks (94%) take a fast path with pure
// pointer-increment addressing; only edge blocks pay for T clamping/masking.
// The 11-tap X-conv + coherence math run on VALU from LDS.
// ---------------------------------------------------------------------------

#define BDIM 16
#define TDIM 2500
#define XDIM 512

typedef __attribute__((ext_vector_type(2))) float v2f;
typedef __attribute__((ext_vector_type(8))) float v8f;

struct CohConsts {
    float ktpad[132];   // zero-padded T taps: ktpad[16+i] = k_t[i], i in [0,100]
    float kx[11];       // X taps
};

#define COH_WMMA5()                                                                                 \
    do {                                                                                            \
        accX  = __builtin_amdgcn_wmma_f32_16x16x4_f32(false, a, false, bX,  (short)0, accX,  false, false); \
        accY  = __builtin_amdgcn_wmma_f32_16x16x4_f32(false, a, false, bY,  (short)0, accY,  false, false); \
        accXX = __builtin_amdgcn_wmma_f32_16x16x4_f32(false, a, false, bXX, (short)0, accXX, false, false); \
        accYY = __builtin_amdgcn_wmma_f32_16x16x4_f32(false, a, false, bYY, (short)0, accYY, false, false); \
        accXY = __builtin_amdgcn_wmma_f32_16x16x4_f32(false, a, false, bXY, (short)0, accXY, false, false); \
    } while (0)

__global__ __launch_bounds__(320)
void LocalCoherence_48180943126735_kernel(const float* __restrict__ x,
                                          const float* __restrict__ y,
                                          float* __restrict__ out,
                                          CohConsts cc, float invCount) {
    __shared__ float sKt[132];
    __shared__ float sKx[11];
    __shared__ float sSm[5][16][162];   // T-smoothed tiles, stride 162 (bank-friendly)
    __shared__ float sRed[10];

    const int tid  = threadIdx.x;
    const int lane = tid & 31;
    const int w    = tid >> 5;                 // wave id 0..9
    const int b    = blockIdx.z;               // batch
    const int tb0  = blockIdx.y * 16;          // output T-row base of this block
    const int xb   = blockIdx.x * 128;         // output X-col base of this block

    // stage taps into LDS
    for (int i = tid; i < 132; i += 320) sKt[i] = cc.ktpad[i];
    if (tid < 11) sKx[tid] = cc.kx[tid];
    __syncthreads();

    // ---------------- Phase 1: 101-tap T-conv via WMMA (per-wave 16x16 tile)
    const int m        = lane & 15;            // A-row (t) / B-col (x) index
    const int kh       = (lane >> 4) << 1;     // K half: 0 or 2
    const int colLocal = w * 16 + m;           // 0..159 (covers xb-16 .. xb+143)
    const int col      = xb - 16 + colLocal;
    const bool colOK   = (col >= 0) && (col < XDIM);
    const int   colC   = colOK ? col : 0;
    const float colM   = colOK ? 1.0f : 0.0f;  // zero-pad (SAME) in X
    const size_t plane = (size_t)b * TDIM * XDIM;

    v8f accX = {}, accY = {}, accXX = {}, accYY = {}, accXY = {};

    if (tb0 >= 50 && tb0 + 65 < TDIM) {
        // ---- fast interior path: no T clamping, pure offset increments ----
        size_t o0 = plane + (size_t)(tb0 - 50 + kh) * XDIM + colC;
        int ai = 16 + kh - m;                  // banded weight index, += 4/chunk
        for (int c = 0; c < 29; ++c) {
            const float xv0 = x[o0] * colM;
            const float xv1 = x[o0 + XDIM] * colM;
            const float yv0 = y[o0] * colM;
            const float yv1 = y[o0 + XDIM] * colM;

            v2f a;  a.x = sKt[ai];        a.y = sKt[ai + 1];
            v2f bX;  bX.x  = xv0;         bX.y  = xv1;
            v2f bY;  bY.x  = yv0;         bY.y  = yv1;
            v2f bXX; bXX.x = xv0 * xv0;   bXX.y = xv1 * xv1;
            v2f bYY; bYY.x = yv0 * yv0;   bYY.y = yv1 * yv1;
            v2f bXY; bXY.x = xv0 * yv0;   bXY.y = xv1 * yv1;

            COH_WMMA5();

            o0 += (size_t)(4 * XDIM);
            ai += 4;
        }
    } else {
        // ---- boundary path: clamp + mask for zero (SAME) padding in T ----
        for (int c = 0; c < 29; ++c) {
            const int r0 = tb0 - 50 + 4 * c + kh;
            const int r1 = r0 + 1;
            const float m0 = (r0 >= 0 && r0 < TDIM) ? colM : 0.0f;
            const float m1 = (r1 >= 0 && r1 < TDIM) ? colM : 0.0f;
            const int rc0 = (r0 < 0) ? 0 : ((r0 >= TDIM) ? (TDIM - 1) : r0);
            const int rc1 = (r1 < 0) ? 0 : ((r1 >= TDIM) ? (TDIM - 1) : r1);
            const size_t o0 = plane + (size_t)rc0 * XDIM + colC;
            const size_t o1 = plane + (size_t)rc1 * XDIM + colC;

            const float xv0 = x[o0] * m0, xv1 = x[o1] * m1;
            const float yv0 = y[o0] * m0, yv1 = y[o1] * m1;

            const int ai = 16 + 4 * c + kh - m;   // in [1, 131]
            v2f a;  a.x = sKt[ai];        a.y = sKt[ai + 1];
            v2f bX;  bX.x  = xv0;         bX.y  = xv1;
            v2f bY;  bY.x  = yv0;         bY.y  = yv1;
            v2f bXX; bXX.x = xv0 * xv0;   bXX.y = xv1 * xv1;
            v2f bYY; bYY.x = yv0 * yv0;   bYY.y = yv1 * yv1;
            v2f bXY; bXY.x = xv0 * yv0;   bXY.y = xv1 * yv1;

            COH_WMMA5();
        }
    }

    // spill D tiles: lane holds col n=lane&15, rows r + 8*(lane>=16)
    {
        const int rowOfs = (lane >> 4) << 3;
#pragma unroll
        for (int r = 0; r < 8; ++r) {
            const int row = r + rowOfs;
            sSm[0][row][colLocal] = accX[r];
            sSm[1][row][colLocal] = accY[r];
            sSm[2][row][colLocal] = accXX[r];
            sSm[3][row][colLocal] = accYY[r];
            sSm[4][row][colLocal] = accXY[r];
        }
    }
    __syncthreads();

    // ---------------- Phase 2: 11-tap X-conv + coherence + partial mean
    float partial = 0.0f;
    for (int idx = tid; idx < 16 * 128; idx += 320) {
        const int row = idx >> 7;
        const int cx  = idx & 127;
        if (tb0 + row < TDIM) {
            float mx = 0.f, my = 0.f, mxx = 0.f, myy = 0.f, mxy = 0.f;
#pragma unroll
            for (int j = 0; j < 11; ++j) {
                const float kw = sKx[j];
                const int ci = cx + 11 + j;    // LDS col of global (xb+cx)+j-5
                mx  += kw * sSm[0][row][ci];
                my  += kw * sSm[1][row][ci];
                mxx += kw * sSm[2][row][ci];
                myy += kw * sSm[3][row][ci];
                mxy += kw * sSm[4][row][ci];
            }
            const float vx  = mxx - mx * mx;
            const float vy  = myy - my * my;
            const float cov = mxy - mx * my;
            const float cs  = cov * __frsqrt_rn(vx * vy + 1e-9f);
            partial += 1.0f - cs;
        }
    }

    // wave reduction -> per-wave LDS slot -> single atomic per block
#pragma unroll
    for (int o = 16; o > 0; o >>= 1) partial += __shfl_down(partial, o, 32);
    if (lane == 0) sRed[w] = partial;
    __syncthreads();
    if (tid == 0) {
        float s = 0.f;
#pragma unroll
        for (int i = 0; i < 10; ++i) s += sRed[i];
        atomicAdd(out, s * invCount);
    }
}

static void coh_fill_consts(CohConsts& cc) {
    double kt[101]; double st = 0.0;
    for (int i = 0; i < 101; ++i) {
        const double d = (double)i - 50.0;
        kt[i] = exp(-(d * d) / (2.0 * 21.0 * 21.0));
        st += kt[i];
    }
    for (int i = 0; i < 132; ++i) cc.ktpad[i] = 0.0f;
    for (int i = 0; i < 101; ++i) cc.ktpad[16 + i] = (float)(kt[i] / st);

    double kx[11]; double sx = 0.0;
    for (int i = 0; i < 11; ++i) {
        const double d = (double)i - 5.0;
        kx[i] = exp(-(d * d) / (2.0 * 11.0 * 11.0));
        sx += kx[i];
    }
    for (int i = 0; i < 11; ++i) cc.kx[i] = (float)(kx[i] / sx);
}

extern "C" void kernel_launch(void* const* d_in, const int* in_sizes, int n_in,
                              void* d_out, int out_size, void* d_ws, size_t ws_size,
                              hipStream_t stream) {
    (void)in_sizes; (void)n_in; (void)d_ws; (void)ws_size;
    const float* x = (const float*)d_in[0];
    const float* y = (const float*)d_in[1];
    float* out = (float*)d_out;

    hipMemsetAsync(out, 0, sizeof(float) * (size_t)out_size, stream);

    CohConsts cc;
    coh_fill_consts(cc);

    const float invCount = 1.0f / ((float)BDIM * (float)TDIM * (float)XDIM);
    dim3 grid(XDIM / 128, (TDIM + 15) / 16, BDIM);   // (4, 157, 16)
    dim3 block(320, 1, 1);                           // 10 waves of 32
    LocalCoherence_48180943126735_kernel<<<grid, block, 0, stream>>>(x, y, out, cc, invCount);
}